// MPLayer_in_K_15831249453135
// MI455X (gfx1250) — compile-verified
//
#include <hip/hip_runtime.h>
#include <cstdint>

// Problem constants (from reference): B=128, IN=1024, OUT=512, GAMMA=32.
// out[b,o] = mean(smallest 32 of zPlus[b,:,o]) - mean(smallest 32 of zMinus[b,:,o])
// zPlus  set over i: { relu(3+x)+relu(3+w), relu(3-x)+relu(3-w) }
// zMinus set over i: { relu(3+x)+relu(3-w), relu(3-x)+relu(3+w) }

#define BDIM   128
#define INDIM  1024
#define OUTDIM 512
#define TB     16      // b-tile per block
#define TO     16      // o-tile per block
#define KC     64      // i-chunk staged per step
#define NCHUNK (INDIM / KC)   // 16

#if defined(__AMDGCN__) && __has_builtin(__builtin_amdgcn_global_load_async_to_lds_b128)
#define HAVE_ASYNC_LDS 1
#else
#define HAVE_ASYNC_LDS 0
#endif

#if defined(__AMDGCN__)
typedef int v4i __attribute__((ext_vector_type(4)));
typedef __attribute__((address_space(1))) v4i* gv4i_p;  // global
typedef __attribute__((address_space(3))) v4i* lv4i_p;  // LDS
// Generic->AS casts must go through an integer in HIP clang.
// Global: generic VA == AS1 VA. LDS: generic VA low 32 bits == LDS offset
// (per the ISA's flat addressing rules).
#define AS_GLOBAL(p) ((gv4i_p)(uintptr_t)(p))
#define AS_LDS(p)    ((lv4i_p)(uint32_t)(uintptr_t)(p))
#endif

__device__ __forceinline__ void wait_async_le2() {
#if defined(__AMDGCN__)
#if __has_builtin(__builtin_amdgcn_s_wait_asynccnt)
  __builtin_amdgcn_s_wait_asynccnt(2);
#else
  asm volatile("s_wait_asynccnt 2" ::: "memory");
#endif
#endif
}

__device__ __forceinline__ void wait_async_le0() {
#if defined(__AMDGCN__)
#if __has_builtin(__builtin_amdgcn_s_wait_asynccnt)
  __builtin_amdgcn_s_wait_asynccnt(0);
#else
  asm volatile("s_wait_asynccnt 0" ::: "memory");
#endif
#endif
}

// Branch-free exact sorted insert into ascending r[0..31], dropping the largest.
// If r[k-1] <= v < r[k]: entries k..30 shift up, v lands at slot k, r[31] drops.
// 63 v_min/v_max ops; the max() terms are all independent and the plain
// fminf/fmaxf forms VOPD dual-issue (v_dual_min/max) — measured lowest static
// instruction count of all insertion formulations tried.
__device__ __forceinline__ void insert32(float (&r)[32], float v) {
#pragma unroll
  for (int j = 31; j >= 1; --j)
    r[j] = fminf(r[j], fmaxf(r[j - 1], v));
  r[0] = fminf(r[0], v);
}

// Stage one chunk: x-tile (TB x KC, row stride INDIM) and w-tile (KC x TO,
// row stride OUTDIM) into LDS. 256 lanes x 16B each per tile (1024 floats).
__device__ __forceinline__ void stage_chunk(const float* __restrict__ xg,
                                            const float* __restrict__ wg,
                                            float* xl, float* wl, int tid) {
  const int xrow = tid >> 4, xseg = tid & 15;  // 16 rows * 16 segs(16B)
  const int wrow = tid >> 2, wseg = tid & 3;   // 64 rows * 4 segs(16B)
  const float* xsrc = xg + xrow * INDIM + xseg * 4;
  const float* wsrc = wg + wrow * OUTDIM + wseg * 4;
  float* xdst = xl + xrow * KC + xseg * 4;
  float* wdst = wl + wrow * TO + wseg * 4;
#if HAVE_ASYNC_LDS
  __builtin_amdgcn_global_load_async_to_lds_b128(AS_GLOBAL(xsrc), AS_LDS(xdst), 0, 0);
  __builtin_amdgcn_global_load_async_to_lds_b128(AS_GLOBAL(wsrc), AS_LDS(wdst), 0, 0);
#else
  *(float4*)xdst = *(const float4*)xsrc;
  *(float4*)wdst = *(const float4*)wsrc;
#endif
}

__global__ __launch_bounds__(TB * TO) void spike_topk_kernel(
    const float* __restrict__ xg,   // (B, IN)
    const float* __restrict__ wg,   // (IN, OUT)
    float* __restrict__ out) {      // (B, OUT)
  __shared__ float xs[2][TB * KC];  // 4 KB per buffer
  __shared__ float wsh[2][KC * TO]; // 4 KB per buffer

  const int tid = threadIdx.x;
  const int o0 = blockIdx.x * TO;
  const int b0 = blockIdx.y * TB;
  const int tx = tid & (TO - 1);   // o within tile
  const int ty = tid >> 4;         // b within tile

  const float* xbase = xg + b0 * INDIM;
  const float* wbase = wg + o0;

  float rP[32], rM[32];
#pragma unroll
  for (int j = 0; j < 32; ++j) { rP[j] = 3.0e38f; rM[j] = 3.0e38f; }

  // Prologue: stage chunk 0 into buffer 0.
  stage_chunk(xbase, wbase, xs[0], wsh[0], tid);

  for (int c = 0; c < NCHUNK; ++c) {
    const int cur = c & 1;
    if (c + 1 < NCHUNK) {
      // Issue next chunk's async loads into the other buffer, then wait for
      // this chunk's pair to land (async loads complete in order: leaving the
      // newest 2 outstanding means chunk c's 2 are done).
      stage_chunk(xbase + (c + 1) * KC, wbase + (c + 1) * KC * OUTDIM,
                  xs[cur ^ 1], wsh[cur ^ 1], tid);
      wait_async_le2();
    } else {
      wait_async_le0();
    }
    __syncthreads();   // every wave waited for its own issues -> tile complete

    const float* xr = &xs[cur][ty * KC];  // broadcast reads across tx
    const float* wr = &wsh[cur][tx];      // 16 consecutive banks, broadcast across ty
    for (int k = 0; k < KC; ++k) {
      const float xv = xr[k];
      const float wv = wr[k * TO];
      const float ap = fmaxf(3.0f + xv, 0.0f);
      const float am = fmaxf(3.0f - xv, 0.0f);
      const float wp = fmaxf(3.0f + wv, 0.0f);
      const float wm = fmaxf(3.0f - wv, 0.0f);
      insert32(rP, ap + wp);
      insert32(rP, am + wm);
      insert32(rM, ap + wm);
      insert32(rM, am + wp);
    }
    __syncthreads();   // all readers done before next iter overwrites 'cur'
  }

  float sp = 0.0f, sm = 0.0f;
#pragma unroll
  for (int j = 0; j < 32; ++j) { sp += rP[j]; sm += rM[j]; }
  out[(b0 + ty) * OUTDIM + (o0 + tx)] = (sp - sm) * (1.0f / 32.0f);
}

extern "C" void kernel_launch(void* const* d_in, const int* in_sizes, int n_in,
                              void* d_out, int out_size, void* d_ws, size_t ws_size,
                              hipStream_t stream) {
  (void)in_sizes; (void)n_in; (void)d_ws; (void)ws_size; (void)out_size;
  const float* x = (const float*)d_in[0];   // (128, 1024) f32
  const float* w = (const float*)d_in[1];   // (1024, 512) f32
  float* out = (float*)d_out;               // (128, 512) f32
  dim3 grid(OUTDIM / TO, BDIM / TB);        // 32 x 8 blocks
  dim3 block(TB * TO);                      // 256 threads = 8 wave32
  hipLaunchKernelGGL(spike_topk_kernel, grid, block, 0, stream, x, w, out);
}